// Router_403726926488
// MI455X (gfx1250) — compile-verified
//
#include <hip/hip_runtime.h>
#include <cstddef>

typedef __attribute__((ext_vector_type(2))) float v2f;
typedef __attribute__((ext_vector_type(8))) float v8f;

#define IN_CAPS 1152
#define OUT_CAPS 10
#define IN_D 8
#define OUT_D 16
#define NB 256
#define NCOL (OUT_CAPS * OUT_D) /* 160 */
#define RT_THREADS 640

// ---------------------------------------------------------------------------
// Kernel 1: u_hat[b,i,j,m] = sum_n W[i,j,n,m] * u[b,i,n]
// One wave computes a 16(batch) x 16(out_d) tile for one (i, j) via two
// V_WMMA_F32_16X16X4_F32 (K = 8 split into 2 x 4).
// Block = 320 threads (10 waves, wave w -> j = w); grid = IN_CAPS * 16 btiles.
// ---------------------------------------------------------------------------
__global__ __launch_bounds__(320)
void uhat_wmma_kernel(const float* __restrict__ u, const float* __restrict__ W,
                      float* __restrict__ uhat) {
  const int tid  = threadIdx.x;
  const int lane = tid & 31;
  const int j    = tid >> 5;              // 0..9  (out capsule)
  const int i    = blockIdx.x >> 4;       // in capsule
  const int b0   = (blockIdx.x & 15) << 4; // batch tile base

  // A-matrix (16x4 fp32): lanes 0-15 -> M=0..15 (K=0,1), lanes 16-31 -> K=2,3
  const int M  = lane & 15;
  const int kb = (lane >> 4) << 1;        // 0 or 2

  const float* up = u + ((size_t)(b0 + M) * IN_CAPS + i) * IN_D;
  v2f a0; a0.x = up[kb];     a0.y = up[kb + 1];       // K-chunk 0: n = 0..3
  v2f a1; a1.x = up[4 + kb]; a1.y = up[5 + kb];       // K-chunk 1: n = 4..7

  // B-matrix (4x16 fp32): row (K) striped across lanes within a VGPR
  const int N = lane & 15;
  const float* wp = W + (((size_t)i * OUT_CAPS + j) * IN_D) * OUT_D;
  v2f bv0; bv0.x = wp[kb * OUT_D + N];       bv0.y = wp[(kb + 1) * OUT_D + N];
  v2f bv1; bv1.x = wp[(4 + kb) * OUT_D + N]; bv1.y = wp[(5 + kb) * OUT_D + N];

  v8f c = {};
  c = __builtin_amdgcn_wmma_f32_16x16x4_f32(false, a0, false, bv0,
                                            (short)0, c, false, false);
  c = __builtin_amdgcn_wmma_f32_16x16x4_f32(false, a1, false, bv1,
                                            (short)0, c, false, false);

  // D layout: VGPR r, lanes 0-15 -> M=r, lanes 16-31 -> M=r+8, N = lane&15
  const int rbase = (lane >> 4) << 3;     // 0 or 8
  const size_t rowStride = (size_t)IN_CAPS * NCOL;
  float* op = uhat + ((size_t)b0 * IN_CAPS + i) * NCOL + j * OUT_D + N;
#pragma unroll
  for (int r = 0; r < 8; ++r) {
    op[(size_t)(rbase + r) * rowStride] = c[r];
  }
}

// ---------------------------------------------------------------------------
// Kernel 2: dynamic routing, one workgroup per batch element.
// Logits b[1152][10] held in LDS across all 3 iterations; u_hat streamed
// from L2 (5 passes total: s+a, s+a, s).
// ---------------------------------------------------------------------------
__global__ __launch_bounds__(RT_THREADS)
void routing_kernel(const float* __restrict__ uhat, float* __restrict__ out) {
  __shared__ float bL[IN_CAPS * OUT_CAPS];   // 46080 B logits
  __shared__ float part[4 * NCOL];           // s-pass partials
  __shared__ float pmax[64 * OUT_CAPS];      // softmax max partials
  __shared__ float psum[64 * OUT_CAPS];      // softmax sum partials
  __shared__ float colmax[OUT_CAPS], colsum[OUT_CAPS];
  __shared__ float sArr[NCOL], vArr[NCOL], scl[OUT_CAPS];

  const int t  = threadIdx.x;
  const int bb = blockIdx.x;
  const float* U = uhat + (size_t)bb * IN_CAPS * NCOL;

  for (int idx = t; idx < IN_CAPS * OUT_CAPS; idx += RT_THREADS) bL[idx] = 0.0f;
  __syncthreads();

  for (int it = 0; it < 3; ++it) {
    // ---- softmax stats over i, per column j ----
    {
      const int j  = t % OUT_CAPS;
      const int sl = t / OUT_CAPS;           // 0..63
      float m = -3.0e38f;
      for (int i = sl; i < IN_CAPS; i += 64) m = fmaxf(m, bL[i * OUT_CAPS + j]);
      pmax[sl * OUT_CAPS + j] = m;
    }
    __syncthreads();
    if (t < OUT_CAPS) {
      float m = -3.0e38f;
      for (int s = 0; s < 64; ++s) m = fmaxf(m, pmax[s * OUT_CAPS + t]);
      colmax[t] = m;
    }
    __syncthreads();
    {
      const int j  = t % OUT_CAPS;
      const int sl = t / OUT_CAPS;
      const float m = colmax[j];
      float s = 0.0f;
      for (int i = sl; i < IN_CAPS; i += 64) s += __expf(bL[i * OUT_CAPS + j] - m);
      psum[sl * OUT_CAPS + j] = s;
    }
    __syncthreads();
    if (t < OUT_CAPS) {
      float s = 0.0f;
      for (int k = 0; k < 64; ++k) s += psum[k * OUT_CAPS + t];
      colsum[t] = s;
    }
    __syncthreads();

    // ---- s[j,m] = sum_i c_ij * uhat[i,j,m]  (4 slices x 160 columns) ----
    {
      const int n  = t % NCOL;
      const int sl = t / NCOL;               // 0..3
      const int j  = n / OUT_D;
      const float m   = colmax[j];
      const float inv = 1.0f / colsum[j];
      float acc = 0.0f;
      for (int i = sl; i < IN_CAPS; i += 4) {
        const float cij = __expf(bL[i * OUT_CAPS + j] - m) * inv;
        acc += cij * U[(size_t)i * NCOL + n];
      }
      part[sl * NCOL + n] = acc;
    }
    __syncthreads();
    if (t < NCOL)
      sArr[t] = part[t] + part[NCOL + t] + part[2 * NCOL + t] + part[3 * NCOL + t];
    __syncthreads();

    // ---- squash ----
    if (t < OUT_CAPS) {
      float s2 = 0.0f;
      for (int m2 = 0; m2 < OUT_D; ++m2) {
        const float x = sArr[t * OUT_D + m2];
        s2 += x * x;
      }
      scl[t] = s2 / ((1.0f + s2) * sqrtf(s2 + 1e-8f));
    }
    __syncthreads();
    if (t < NCOL) vArr[t] = scl[t / OUT_D] * sArr[t];
    __syncthreads();

    if (it == 2) break;  // last iteration: a-pass / logit update are dead

    // ---- b_ij += sum_m v[j,m] * uhat[i,j,m] ----
    for (int idx = t; idx < IN_CAPS * OUT_CAPS; idx += RT_THREADS) {
      const int i = idx / OUT_CAPS;
      const int j = idx % OUT_CAPS;
      const float* p = U + (size_t)i * NCOL + j * OUT_D;
      float acc = 0.0f;
#pragma unroll
      for (int m2 = 0; m2 < OUT_D; ++m2) acc += vArr[j * OUT_D + m2] * p[m2];
      bL[idx] += acc;
    }
    __syncthreads();
  }

  if (t < NCOL) out[(size_t)bb * NCOL + t] = vArr[t];
}

extern "C" void kernel_launch(void* const* d_in, const int* in_sizes, int n_in,
                              void* d_out, int out_size, void* d_ws, size_t ws_size,
                              hipStream_t stream) {
  (void)in_sizes; (void)n_in; (void)out_size; (void)ws_size;
  const float* u = (const float*)d_in[0];   // [256,1152,8]
  const float* W = (const float*)d_in[1];   // [1152,10,8,16]
  float* uhat = (float*)d_ws;               // [256,1152,10,16] fp32 (~189 MB)

  uhat_wmma_kernel<<<IN_CAPS * 16, 320, 0, stream>>>(u, W, uhat);
  routing_kernel<<<NB, RT_THREADS, 0, stream>>>(uhat, (float*)d_out);
}